// CapsuleLayer_4372276707524
// MI455X (gfx1250) — compile-verified
//
#include <hip/hip_runtime.h>
#include <stdint.h>

// CapsNet dynamic routing, fully fused: pondered + 3 routing iterations + transpose,
// one pass over the 188.7 MB output. No matmul exists in this op -> WMMA not applicable;
// CDNA5 async global->LDS (ASYNCcnt) path used to broadcast the per-u weight slice.

#define ROUTING_ITERS 3
#define EPS_F 1e-8f

namespace {
constexpr int B = 32, C = 8, U = 1152, N = 10, D = 16;
constexpr int NC      = N * C;          // 80
constexpr int GROUPS  = B * NC;         // 2560 independent 16-vectors per u
constexpr int THREADS = 256;            // 8 wave32 per block
constexpr int GPT     = GROUPS / THREADS; // 10 groups per thread
constexpr int WFLOATS = N * C * D;      // 1280 floats of weights per u
constexpr int WCHUNKS = WFLOATS / 4;    // 320 x 16-byte async chunks
}

// Async copy of one 16B chunk global->LDS. gfx1250: tracked with ASYNCcnt.
// Generic LDS pointer low 32 bits == wave-relative LDS offset (flat LDS aperture
// truncation, ISA 10.2), which is exactly what VDST of the async op expects.
__device__ __forceinline__ void async_copy_b128(const float* gsrc, float* lds_dst) {
  uint32_t la = (uint32_t)(uintptr_t)lds_dst;       // LDS byte offset
  uint64_t ga = (uint64_t)(uintptr_t)gsrc;          // 64-bit global address
  asm volatile("global_load_async_to_lds_b128 %0, %1, off"
               :: "v"(la), "v"(ga) : "memory");
}

__device__ __forceinline__ void wait_asynccnt0() {
  asm volatile("s_wait_asynccnt 0" ::: "memory");
}

__global__ __launch_bounds__(THREADS)
void caps_route_kernel(const float* __restrict__ in,   // (B, C, U)
                       const float* __restrict__ w,    // (U, N, C, D)
                       float* __restrict__ out)        // (B, N, U, C, D)
{
  __shared__ __align__(16) float wlds[WFLOATS]; // 5120 B: weights[u,:,:,:]
  __shared__ float xlds[B * C];                 // 1024 B: inputs[:,:,u]

  const int u = blockIdx.x;
  const int t = threadIdx.x;

  // Stage inputs: thread t = b*8+c loads inputs[b][c][u]
  xlds[t] = in[t * U + u];

  // Async-stage the 5120B weight slice for this u (320 b128 chunks / 256 threads)
  const float* wsrc = w + (size_t)u * WFLOATS;
  async_copy_b128(wsrc + t * 4, wlds + t * 4);
  if (t < WCHUNKS - THREADS)  // remaining 64 chunks
    async_copy_b128(wsrc + (THREADS + t) * 4, wlds + (THREADS + t) * 4);
  wait_asynccnt0();
  __syncthreads();

  const float LOG2E = 1.4426950408889634f;

  #pragma unroll 1
  for (int i = 0; i < GPT; ++i) {
    const int g  = t + THREADS * i;   // 0..2559
    const int b  = g / NC;
    const int nc = g - b * NC;        // n*8 + c
    const int cc = nc & 7;

    const float x = xlds[(b << 3) | cc];

    // p[d] = weights[u,n,c,d] * inputs[b,c,u]   (ds_load_b128 x4)
    float p[D];
    const float4* wv = (const float4*)(wlds + nc * D);
    {
      float4 w0 = wv[0], w1 = wv[1], w2 = wv[2], w3 = wv[3];
      p[0]=w0.x*x;  p[1]=w0.y*x;  p[2]=w0.z*x;  p[3]=w0.w*x;
      p[4]=w1.x*x;  p[5]=w1.y*x;  p[6]=w1.z*x;  p[7]=w1.w*x;
      p[8]=w2.x*x;  p[9]=w2.y*x;  p[10]=w2.z*x; p[11]=w2.w*x;
      p[12]=w3.x*x; p[13]=w3.y*x; p[14]=w3.z*x; p[15]=w3.w*x;
    }

    float l2[D];  // logits, pre-scaled by log2(e) so v_exp_f32 applies directly
    float o[D];

    // ---- iteration 1: logits==0 -> softmax is exactly uniform 1/16 (no exp needed)
    {
      float sq = 0.f;
      #pragma unroll
      for (int d = 0; d < D; ++d) { float s = p[d] * 0.0625f; o[d] = s; sq = fmaf(s, s, sq); }
      float scale = sq * __builtin_amdgcn_rcpf(1.0f + sq) * __builtin_amdgcn_rsqf(sq + EPS_F);
      #pragma unroll
      for (int d = 0; d < D; ++d) { o[d] *= scale; l2[d] = (p[d] * o[d]) * LOG2E; }
    }

    // ---- iterations 2..3: real softmax over d, squash, logits update
    #pragma unroll
    for (int it = 1; it < ROUTING_ITERS; ++it) {
      float m = l2[0];
      #pragma unroll
      for (int d = 1; d < D; ++d) m = fmaxf(m, l2[d]);
      float e[D]; float sum = 0.f;
      #pragma unroll
      for (int d = 0; d < D; ++d) { e[d] = __builtin_amdgcn_exp2f(l2[d] - m); sum += e[d]; }
      const float inv = __builtin_amdgcn_rcpf(sum);
      float sq = 0.f;
      #pragma unroll
      for (int d = 0; d < D; ++d) { float s = (e[d] * inv) * p[d]; e[d] = s; sq = fmaf(s, s, sq); }
      float scale = sq * __builtin_amdgcn_rcpf(1.0f + sq) * __builtin_amdgcn_rsqf(sq + EPS_F);
      #pragma unroll
      for (int d = 0; d < D; ++d) {
        float oo = e[d] * scale;
        o[d] = oo;
        l2[d] = fmaf(p[d] * oo, LOG2E, l2[d]);   // logits += pondered*out (log2-domain)
      }
    }

    // store out[b][n][u][c][0..15] : 64 contiguous bytes, 16B-aligned
    const int n = nc >> 3;
    float* op = out + ((size_t)(b * N + n) * U + u) * (C * D) + cc * D;
    float4* ov = (float4*)op;
    ov[0] = make_float4(o[0],  o[1],  o[2],  o[3]);
    ov[1] = make_float4(o[4],  o[5],  o[6],  o[7]);
    ov[2] = make_float4(o[8],  o[9],  o[10], o[11]);
    ov[3] = make_float4(o[12], o[13], o[14], o[15]);
  }
}

extern "C" void kernel_launch(void* const* d_in, const int* in_sizes, int n_in,
                              void* d_out, int out_size, void* d_ws, size_t ws_size,
                              hipStream_t stream) {
  (void)in_sizes; (void)n_in; (void)out_size; (void)d_ws; (void)ws_size;
  const float* in = (const float*)d_in[0];   // (32, 8, 1152) f32
  const float* w  = (const float*)d_in[1];   // (1152, 10, 8, 16) f32
  float* out      = (float*)d_out;           // (32, 10, 1152, 8, 16) f32
  caps_route_kernel<<<dim3(U), dim3(THREADS), 0, stream>>>(in, w, out);
}